// SetConv_51247549776076
// MI455X (gfx1250) — compile-verified
//
#include <hip/hip_runtime.h>

// SetConv (PointNet++ set abstraction) for MI455X / gfx1250, wave32.
// GEMM layers use V_WMMA_F32_16X16X4_F32 (exact f32, matches f32 reference).
// All reductions deterministic (no float atomics) for graph-replay stability.

typedef float v2f __attribute__((ext_vector_type(2)));
typedef float v8f __attribute__((ext_vector_type(8)));

constexpr int NB   = 4;            // batches
constexpr int NPTS = 8192;         // input points
constexpr int NC   = 64;           // input feature channels
constexpr int NM   = 1024;         // sampled points
constexpr int NK   = 32;           // neighbors per sample
constexpr int NP   = NB * NM * NK; // 131072 "pixels" (power of 2: 2^17)
constexpr float RADIUS2 = 0.25f;
constexpr float BN_EPS  = 1e-3f;
constexpr int C0 = 68;             // 3 + NC, padded to multiple of 4 (WMMA K step)

__device__ inline unsigned ballot32(bool p) {
#if __has_builtin(__builtin_amdgcn_ballot_w32)
  return __builtin_amdgcn_ballot_w32(p);
#else
  return (unsigned)__ballot(p ? 1 : 0);
#endif
}

// ---------------- 1) Farthest point sampling: one block per batch ----------
__global__ __launch_bounds__(1024) void fps_kernel(const float* __restrict__ points,
                                                   float* __restrict__ newpts) {
  const int b = blockIdx.x;
  const int t = threadIdx.x;
  constexpr int T = 1024;
  constexpr int PPT = NPTS / T; // 8 points per thread
  const float* pb = points + (size_t)b * NPTS * 3;

  float px[PPT], py[PPT], pz[PPT], dmin[PPT];
#pragma unroll
  for (int i = 0; i < PPT; ++i) {
    const int n = i * T + t;
    px[i] = pb[3 * n + 0];
    py[i] = pb[3 * n + 1];
    pz[i] = pb[3 * n + 2];
    dmin[i] = 1e10f;
  }

  __shared__ float sval[T];
  __shared__ int   sidx[T];

  int last = 0;
  if (t == 0) {
    newpts[((size_t)b * NM) * 3 + 0] = pb[0];
    newpts[((size_t)b * NM) * 3 + 1] = pb[1];
    newpts[((size_t)b * NM) * 3 + 2] = pb[2];
  }

  for (int s = 1; s < NM; ++s) {
    const float lx = pb[3 * last + 0];
    const float ly = pb[3 * last + 1];
    const float lz = pb[3 * last + 2];
    float bv = -1.0f;
    int   bi = 0;
#pragma unroll
    for (int i = 0; i < PPT; ++i) {
      const float dx = px[i] - lx, dy = py[i] - ly, dz = pz[i] - lz;
      const float d  = dx * dx + dy * dy + dz * dz;
      const float nd = fminf(dmin[i], d);
      dmin[i] = nd;
      if (nd > bv) { bv = nd; bi = i * T + t; } // strict > keeps lowest n per thread
    }
    sval[t] = bv; sidx[t] = bi;
    __syncthreads();
    for (int off = T / 2; off > 0; off >>= 1) {
      if (t < off) {
        const float ov = sval[t + off];
        const int   oi = sidx[t + off];
        if (ov > sval[t] || (ov == sval[t] && oi < sidx[t])) { sval[t] = ov; sidx[t] = oi; }
      }
      __syncthreads();
    }
    last = sidx[0];
    __syncthreads(); // protect sval/sidx before next round's writes
    if (t == 0) {
      newpts[((size_t)b * NM + s) * 3 + 0] = pb[3 * last + 0];
      newpts[((size_t)b * NM + s) * 3 + 1] = pb[3 * last + 1];
      newpts[((size_t)b * NM + s) * 3 + 2] = pb[3 * last + 2];
    }
  }
}

// ---------------- 2) Ball query: one wave per centroid ---------------------
__global__ __launch_bounds__(32) void ballq_kernel(const float* __restrict__ points,
                                                   const float* __restrict__ newpts,
                                                   int* __restrict__ ball_idx) {
  const int bm   = blockIdx.x;  // b*NM + m
  const int b    = bm / NM;
  const int lane = threadIdx.x;
  const float cx = newpts[(size_t)bm * 3 + 0];
  const float cy = newpts[(size_t)bm * 3 + 1];
  const float cz = newpts[(size_t)bm * 3 + 2];
  const float* pb = points + (size_t)b * NPTS * 3;
  int* out = ball_idx + (size_t)bm * NK;

  float minv = 3.4e38f;
  int   mini = 0;
  int   count = 0;
  int   firstIdx = -1;

  for (int base = 0; base < NPTS; base += 32) {
    const int n = base + lane;
    const float dx = pb[3 * n + 0] - cx;
    const float dy = pb[3 * n + 1] - cy;
    const float dz = pb[3 * n + 2] - cz;
    const float d2 = dx * dx + dy * dy + dz * dz;
    if (d2 < minv) { minv = d2; mini = n; } // strict < keeps lowest n per lane
    const bool inR = d2 < RADIUS2;
    const unsigned mask = ballot32(inR);            // uniform
    if (firstIdx < 0 && mask != 0u)
      firstIdx = base + __builtin_ctz(mask);        // uniform
    const int pos = count + __builtin_popcount(mask & ((1u << lane) - 1u));
    if (inR && pos < NK) out[pos] = n;              // ascending-index selection
    count += __builtin_popcount(mask);              // uniform
    if (count >= NK) break;                         // uniform early exit
  }

  __shared__ float rv[32];
  __shared__ int   ri[32];
  if (count == 0) { // reference fallback: nearest point (first occurrence on ties)
    rv[lane] = minv; ri[lane] = mini;
    __syncthreads();
    for (int off = 16; off > 0; off >>= 1) {
      if (lane < off) {
        const float ov = rv[lane + off];
        const int   oi = ri[lane + off];
        if (ov < rv[lane] || (ov == rv[lane] && oi < ri[lane])) { rv[lane] = ov; ri[lane] = oi; }
      }
      __syncthreads();
    }
    firstIdx = ri[0];
  }
  for (int kk = count + lane; kk < NK; kk += 32) out[kk] = firstIdx; // pad
}

// ---------------- 3) Build x0 [C0][NP]: gather + relative coords -----------
__global__ void build_x0_kernel(const float* __restrict__ points,
                                const float* __restrict__ features,
                                const float* __restrict__ newpts,
                                const int* __restrict__ ball_idx,
                                float* __restrict__ x0) {
  const int p = blockIdx.x * blockDim.x + threadIdx.x;
  if (p >= NP) return;
  const int bm = p / NK;
  const int b  = bm / NM;
  const int n  = ball_idx[p];
  const float* pt = points + ((size_t)b * NPTS + n) * 3;
  const float* cp = newpts + (size_t)bm * 3;
  x0[(size_t)0 * NP + p] = pt[0] - cp[0];
  x0[(size_t)1 * NP + p] = pt[1] - cp[1];
  x0[(size_t)2 * NP + p] = pt[2] - cp[2];
  for (int c = 0; c < NC; ++c)
    x0[(size_t)(3 + c) * NP + p] = features[((size_t)b * NC + c) * NPTS + n];
  x0[(size_t)(C0 - 1) * NP + p] = 0.f; // zero pad channel 67
}

// ---------------- weight pad: w1 [64][67] -> [64][68] ----------------------
__global__ void padw1_kernel(const float* __restrict__ w1, float* __restrict__ wpad) {
  const int i = blockIdx.x * blockDim.x + threadIdx.x;
  if (i >= 64 * C0) return;
  const int r = i / C0, c = i % C0;
  wpad[i] = (c < 67) ? w1[r * 67 + c] : 0.f;
}

// ---------------- 4) GEMM via V_WMMA_F32_16X16X4_F32 -----------------------
// y[o][p] = sum_c w[o][c] * x[c][p] + bias[o]
// grid = (NP/16 pixel tiles, Cout/16 channel tiles), block = 1 wave (32 lanes).
// A frag (16x4 f32): lanes 0-15 -> K=k,k+1 ; lanes 16-31 -> K=k+2,k+3 (row M=lane&15)
// B frag (4x16 f32): same K split across lane halves, N = lane&15
// C/D (v8f): VGPR r -> M = r + (lane<16 ? 0 : 8), N = lane&15
__global__ __launch_bounds__(32) void gemm_wmma_kernel(const float* __restrict__ x,
                                                       const float* __restrict__ w,
                                                       const float* __restrict__ bias,
                                                       float* __restrict__ y,
                                                       int Cin) {
  const int p0   = blockIdx.x * 16;
  const int o0   = blockIdx.y * 16;
  const int lane = threadIdx.x;
  const int l    = lane & 15;
  const int hi2  = (lane >> 4) << 1; // 0 or 2

  v8f acc = {0.f, 0.f, 0.f, 0.f, 0.f, 0.f, 0.f, 0.f};
  for (int k = 0; k < Cin; k += 4) {
    const float* wr = w + (size_t)(o0 + l) * Cin + (k + hi2);
    v2f a = {wr[0], wr[1]};
    const float* xr = x + (size_t)(k + hi2) * NP + (p0 + l);
    v2f bfrag = {xr[0], xr[NP]};
    acc = __builtin_amdgcn_wmma_f32_16x16x4_f32(false, a, false, bfrag,
                                                (short)0, acc, false, false);
  }
  const int orow = o0 + (lane >> 4) * 8;
  const int pcol = p0 + l;
#pragma unroll
  for (int r = 0; r < 8; ++r)
    y[(size_t)(orow + r) * NP + pcol] = acc[r] + bias[orow + r];
}

// ---------------- 5) BatchNorm stats (deterministic two-stage) -------------
__global__ void stats_partial_kernel(const float* __restrict__ y, float* __restrict__ part) {
  const int c     = blockIdx.x;
  const int chunk = blockIdx.y;       // 0..15
  constexpr int NCHUNK = 16;
  constexpr int CH = NP / NCHUNK;     // 8192
  const float* yc = y + (size_t)c * NP + (size_t)chunk * CH;
  float s = 0.f, s2 = 0.f;
  for (int i = threadIdx.x; i < CH; i += 256) {
    const float v = yc[i];
    s += v; s2 += v * v;
  }
  __shared__ float ss[256], sq[256];
  ss[threadIdx.x] = s; sq[threadIdx.x] = s2;
  __syncthreads();
  for (int off = 128; off > 0; off >>= 1) {
    if (threadIdx.x < off) {
      ss[threadIdx.x] += ss[threadIdx.x + off];
      sq[threadIdx.x] += sq[threadIdx.x + off];
    }
    __syncthreads();
  }
  if (threadIdx.x == 0) {
    part[((size_t)c * NCHUNK + chunk) * 2 + 0] = ss[0];
    part[((size_t)c * NCHUNK + chunk) * 2 + 1] = sq[0];
  }
}

__global__ void stats_final_kernel(const float* __restrict__ part,
                                   float* __restrict__ stats, int Cout) {
  const int c = blockIdx.x * blockDim.x + threadIdx.x;
  if (c >= Cout) return;
  float s = 0.f, s2 = 0.f;
  for (int i = 0; i < 16; ++i) {
    s  += part[((size_t)c * 16 + i) * 2 + 0];
    s2 += part[((size_t)c * 16 + i) * 2 + 1];
  }
  stats[2 * c + 0] = s;
  stats[2 * c + 1] = s2;
}

// ---------------- 6) normalize + affine + ReLU (in place) ------------------
__global__ void bnrelu_kernel(float* __restrict__ y, const float* __restrict__ stats,
                              const float* __restrict__ gamma, const float* __restrict__ beta,
                              int Cout) {
  const size_t i = (size_t)blockIdx.x * blockDim.x + threadIdx.x;
  if (i >= (size_t)Cout * NP) return;
  const int c = (int)(i >> 17); // NP == 2^17
  const float inv = 1.f / (float)NP;
  const float mu  = stats[2 * c + 0] * inv;
  const float var = stats[2 * c + 1] * inv - mu * mu;
  const float v = (y[i] - mu) * rsqrtf(var + BN_EPS) * gamma[c] + beta[c];
  y[i] = fmaxf(v, 0.f);
}

// ---------------- 7) max over K neighbors ----------------------------------
__global__ void maxk_kernel(const float* __restrict__ y3, float* __restrict__ outf) {
  const int i = blockIdx.x * blockDim.x + threadIdx.x; // [B][128][M] flat
  if (i >= NB * 128 * NM) return;
  const int m = i % NM;
  const int o = (i / NM) % 128;
  const int b = i / (128 * NM);
  const float* src = y3 + (size_t)o * NP + ((size_t)(b * NM + m)) * NK;
  float v = src[0];
#pragma unroll
  for (int k = 1; k < NK; ++k) v = fmaxf(v, src[k]);
  outf[i] = v;
}

// ---------------- host orchestration ---------------------------------------
extern "C" void kernel_launch(void* const* d_in, const int* in_sizes, int n_in,
                              void* d_out, int out_size, void* d_ws, size_t ws_size,
                              hipStream_t stream) {
  (void)in_sizes; (void)n_in; (void)out_size; (void)ws_size;
  const float* points   = (const float*)d_in[0];
  const float* features = (const float*)d_in[1];
  const float* w1 = (const float*)d_in[2];  const float* b1 = (const float*)d_in[3];
  const float* g1 = (const float*)d_in[4];  const float* be1 = (const float*)d_in[5];
  const float* w2 = (const float*)d_in[6];  const float* b2 = (const float*)d_in[7];
  const float* g2 = (const float*)d_in[8];  const float* be2 = (const float*)d_in[9];
  const float* w3 = (const float*)d_in[10]; const float* b3 = (const float*)d_in[11];
  const float* g3 = (const float*)d_in[12]; const float* be3 = (const float*)d_in[13];

  float* out     = (float*)d_out;
  float* newpts  = out;                       // [B][M][3]
  float* outfeat = out + (size_t)NB * NM * 3; // [B][128][M]

  char* ws = (char*)d_ws;
  size_t off = 0;
  auto alloc = [&](size_t bytes) {
    size_t r = off;
    off = (off + bytes + 255) & ~(size_t)255;
    return r;
  };
  int*   ball_idx = (int*)(ws + alloc((size_t)NP * 4));
  float* stats    = (float*)(ws + alloc(2 * 128 * 4));
  float* part     = (float*)(ws + alloc((size_t)128 * 16 * 2 * 4));
  float* wpad1    = (float*)(ws + alloc((size_t)64 * C0 * 4));
  float* bufA     = (float*)(ws + alloc((size_t)C0 * NP * 4));  // x0, then y2
  float* bufB     = (float*)(ws + alloc((size_t)128 * NP * 4)); // y1, then y3

  fps_kernel<<<NB, 1024, 0, stream>>>(points, newpts);
  ballq_kernel<<<NB * NM, 32, 0, stream>>>(points, newpts, ball_idx);
  padw1_kernel<<<(64 * C0 + 255) / 256, 256, 0, stream>>>(w1, wpad1);
  build_x0_kernel<<<NP / 256, 256, 0, stream>>>(points, features, newpts, ball_idx, bufA);

  // Layer 1: 68(padded) -> 64   (x0 in bufA, y1 in bufB)
  gemm_wmma_kernel<<<dim3(NP / 16, 64 / 16), 32, 0, stream>>>(bufA, wpad1, b1, bufB, C0);
  stats_partial_kernel<<<dim3(64, 16), 256, 0, stream>>>(bufB, part);
  stats_final_kernel<<<1, 128, 0, stream>>>(part, stats, 64);
  bnrelu_kernel<<<64 * (NP / 256), 256, 0, stream>>>(bufB, stats, g1, be1, 64);

  // Layer 2: 64 -> 64           (y1 in bufB, y2 in bufA)
  gemm_wmma_kernel<<<dim3(NP / 16, 64 / 16), 32, 0, stream>>>(bufB, w2, b2, bufA, 64);
  stats_partial_kernel<<<dim3(64, 16), 256, 0, stream>>>(bufA, part);
  stats_final_kernel<<<1, 128, 0, stream>>>(part, stats, 64);
  bnrelu_kernel<<<64 * (NP / 256), 256, 0, stream>>>(bufA, stats, g2, be2, 64);

  // Layer 3: 64 -> 128          (y2 in bufA, y3 in bufB)
  gemm_wmma_kernel<<<dim3(NP / 16, 128 / 16), 32, 0, stream>>>(bufA, w3, b3, bufB, 64);
  stats_partial_kernel<<<dim3(128, 16), 256, 0, stream>>>(bufB, part);
  stats_final_kernel<<<1, 128, 0, stream>>>(part, stats, 128);
  bnrelu_kernel<<<128 * (NP / 256), 256, 0, stream>>>(bufB, stats, g3, be3, 128);

  maxk_kernel<<<(NB * 128 * NM) / 256, 256, 0, stream>>>(bufB, outfeat);
}